// SpatialFeatureExtractor_65764539236415
// MI455X (gfx1250) — compile-verified
//
#include <hip/hip_runtime.h>
#include <cstdint>

#define BATCH   8
#define GRID_N  2048
#define NC      5

#define BX      64            // threads in x
#define BY      4             // rows per block
#define PTS     4             // x-points per thread (one float4)
#define TILE_X  (BX * PTS)    // 256
#define SEG_F   (TILE_X * NC) // 1280 floats per output row segment
#define SEG_F4  (SEG_F / 4)   // 320 float4 per segment (multiple of 32 -> wave-coalesced)

// CDNA5 async store: DMA 16 bytes straight from LDS to global memory.
// Non-temporal store hint: output is write-once/stream -> don't let 671 MB of
// stores evict the (L2-resident, reused) 134 MB input.
__device__ __forceinline__ void async_store_b128_from_lds_nt(const float* gaddr, uint32_t ldsaddr) {
    asm volatile("global_store_async_from_lds_b128 %0, %1, off th:TH_STORE_NT"
                 :: "v"((uint64_t)(uintptr_t)gaddr), "v"(ldsaddr)
                 : "memory");
}

__device__ __forceinline__ void wait_asynccnt0() {
    asm volatile("s_wait_asynccnt 0" ::: "memory");
}

__global__ __launch_bounds__(BX * BY)
void spatial_features_kernel(const float* __restrict__ u, float* __restrict__ out) {
    // One 256x4 output tile staged channel-interleaved: 4 segs * 1280 floats = 20 KB
    __shared__ __align__(16) float sm[BY * SEG_F];

    const int tx = threadIdx.x;
    const int ty = threadIdx.y;
    const int x0 = blockIdx.x * TILE_X;
    const int y0 = blockIdx.y * BY;
    const int b  = blockIdx.z;
    const int y  = y0 + ty;
    const int x  = x0 + tx * PTS;

    const float* rowC = u + ((size_t)b * GRID_N + y) * GRID_N;

    const float4 c4 = *(const float4*)(rowC + x);
    float4 up4 = make_float4(0.f, 0.f, 0.f, 0.f);
    float4 dn4 = make_float4(0.f, 0.f, 0.f, 0.f);
    if (y > 0)          up4 = *(const float4*)(rowC - GRID_N + x);   // row y-1
    if (y < GRID_N - 1) dn4 = *(const float4*)(rowC + GRID_N + x);   // row y+1
    const float lf0 = (x > 0) ? rowC[x - 1] : 0.0f;                  // zero pad
    const float rt3 = (x + PTS < GRID_N) ? rowC[x + PTS] : 0.0f;     // zero pad

    const float c [4] = {c4.x, c4.y, c4.z, c4.w};
    const float up[4] = {up4.x, up4.y, up4.z, up4.w};
    const float dn[4] = {dn4.x, dn4.y, dn4.z, dn4.w};
    const float lf[4] = {lf0,  c4.x, c4.y, c4.z};
    const float rt[4] = {c4.y, c4.z, c4.w, rt3};

    float vals[PTS * NC];
    #pragma unroll
    for (int p = 0; p < PTS; ++p) {
        const float gx = 0.5f * (rt[p] - lf[p]);
        const float gy = 0.5f * (dn[p] - up[p]);
        vals[p * NC + 0] = up[p] + dn[p] + lf[p] + rt[p] - 4.0f * c[p]; // laplacian
        vals[p * NC + 1] = gx;                                          // grad_x
        vals[p * NC + 2] = gy;                                          // grad_y
        vals[p * NC + 3] = c[p] * c[p];                                 // u^2
        vals[p * NC + 4] = sqrtf(gx * gx + gy * gy + 1e-8f);            // |grad|
    }

    // Stage 20 floats (= 5 x ds_store_b128, 16B aligned: tx*80 bytes) into LDS.
    float4* dst = (float4*)&sm[(ty * BX + tx) * (PTS * NC)];
    #pragma unroll
    for (int i = 0; i < 5; ++i)
        dst[i] = make_float4(vals[4*i], vals[4*i+1], vals[4*i+2], vals[4*i+3]);

    __syncthreads();  // all ds writes visible before async engine reads LDS

    // Stream tile out: 1280 float4 total, 256 threads x 5 iterations.
    // Lane-consecutive 16B addresses -> 512B contiguous per wave instruction.
    const int lid = ty * BX + tx;
    #pragma unroll
    for (int i = 0; i < 5; ++i) {
        const int idx    = i * (BX * BY) + lid;      // 0..1279
        const int seg    = idx / SEG_F4;             // output row within tile
        const int within = idx - seg * SEG_F4;       // float4 index in row segment
        const float* gaddr = out
            + (((size_t)b * GRID_N + (y0 + seg)) * GRID_N + x0) * NC
            + (size_t)within * 4;
        const uint32_t laddr = (uint32_t)(uintptr_t)&sm[seg * SEG_F + within * 4];
        async_store_b128_from_lds_nt(gaddr, laddr);
    }

    // LDS must stay live until the async DMA drains (ASYNCcnt -> 0).
    wait_asynccnt0();
}

extern "C" void kernel_launch(void* const* d_in, const int* in_sizes, int n_in,
                              void* d_out, int out_size, void* d_ws, size_t ws_size,
                              hipStream_t stream) {
    const float* u   = (const float*)d_in[0];
    float*       out = (float*)d_out;
    dim3 block(BX, BY);
    dim3 grid(GRID_N / TILE_X, GRID_N / BY, BATCH);  // 8 x 512 x 8
    spatial_features_kernel<<<grid, block, 0, stream>>>(u, out);
}